// C4SwiGLUExecutor_62380105007377
// MI455X (gfx1250) — compile-verified
//
#include <hip/hip_runtime.h>
#include <math.h>

#define VM_N 16777216
#define SCL  20.0f

typedef __attribute__((ext_vector_type(2))) float v2f;
typedef __attribute__((ext_vector_type(8))) float v8f;

__device__ __forceinline__ float sigm(float x){ return 1.0f / (1.0f + __expf(-x)); }
__device__ __forceinline__ float silu_(float x){ return x * sigm(x); }
// silu_threshold with s = 20
__device__ __forceinline__ float sth(float x){
  float d = SCL * x;
  return (silu_(d + 0.5f * SCL) - silu_(d - 0.5f * SCL)) * (1.0f / SCL);
}
__device__ __forceinline__ float eqg(float a, float b){
  float diff = a - b;
  return sth(diff + 0.5f) * sth(-diff + 0.5f);
}
__device__ __forceinline__ float geg(float a, float b){ return sth(a - b + 0.5f); }
__device__ __forceinline__ float gtg(float a, float b){ return sth(a - b - 0.5f); }
__device__ __forceinline__ float smul(float a, float b){ return a * silu_(b) - a * silu_(-b); }

// ---------------------------------------------------------------------------
// Kernel A: single block. WMMA-based windowed soft-attention reads (5 reads,
// numerators + denominators via one 16x16x4xF32 accumulation chain), then
// block-parallel div/shift enumerations, then scalar VM update.
// Emits: out[0..3] = new pc/sp/bp/ax ; ws[0..4]=write addrs, ws[5..9]=write vals
// ---------------------------------------------------------------------------
__global__ __launch_bounds__(256)
void vm_step_scalar(const float* __restrict__ pcp, const float* __restrict__ spp,
                    const float* __restrict__ bpp, const float* __restrict__ axp,
                    const float* __restrict__ mem,
                    float* __restrict__ out, float* __restrict__ ws)
{
  __shared__ float dmat[32][8];   // dumped D-matrix VGPRs
  __shared__ float reads[5];
  __shared__ float sh[2];         // 0: stack_top, 1: ax_safe
  __shared__ float acc[3];        // div, shl, shr partial sums

  const float pc = pcp[0], sp = spp[0], bp = bpp[0], ax = axp[0];
  const int t = threadIdx.x;
  if (t == 0){ acc[0] = 0.f; acc[1] = 0.f; acc[2] = 0.f; }

  // ---- Phase 1: gathers via WMMA (wave 0 only; full wave => EXEC all ones) --
  if (t < 32){
    const int lane = t;
    const int half = lane >> 4;   // K-half selector
    const int lrow = lane & 15;   // row for A, col for B
    // read addresses: r0=pc r1=sp r2=ax r3=bp r4=sp+8
    float a = (lrow == 0) ? pc : (lrow == 1) ? sp : (lrow == 2) ? ax
            : (lrow == 3) ? bp : (sp + 8.0f);
    float cf = fminf(fmaxf(rintf(a), 0.0f), (float)(VM_N - 1));
    long ctr = (long)cf;

    v8f c = {0.f,0.f,0.f,0.f,0.f,0.f,0.f,0.f};
    #pragma unroll
    for (int ck = 0; ck < 5; ++ck){           // K = 20 total (window 19 + pad)
      float a0 = 0.f, a1 = 0.f, b0 = 0.f, b1 = 0.f;
      const int k0 = 4*ck + 2*half;
      if (lrow < 5){
        #pragma unroll
        for (int j = 0; j < 2; ++j){
          int kk = k0 + j;
          if (kk < 19){
            long pos = ctr - 9 + kk;
            if (pos >= 0 && pos < VM_N){
              float w = eqg(a, (float)pos);   // gate weight (A row lrow)
              float m = mem[pos];             // memory value (B col lrow)
              if (j == 0){ a0 = w; b0 = m; } else { a1 = w; b1 = m; }
            }
          }
        }
      } else if (lrow == 8){
        b0 = 1.0f; b1 = 1.0f;                 // ones column -> denominators
      }
      v2f A = { a0, a1 };
      v2f B = { b0, b1 };
      // D[r][r] = sum_k w_r*mem_r  ;  D[r][8] = sum_k w_r
      c = __builtin_amdgcn_wmma_f32_16x16x4_f32(
            false, A, false, B, (short)0, c, false, false);
    }
    #pragma unroll
    for (int j = 0; j < 8; ++j) dmat[lane][j] = c[j];
  }
  __syncthreads();

  if (t == 0){
    #pragma unroll
    for (int r = 0; r < 5; ++r){
      float num = dmat[r][r];    // M=r,N=r   -> VGPR r, lane r
      float den = dmat[8][r];    // M=r,N=8   -> VGPR r, lane 8
      reads[r] = num / (den + 1e-8f);
    }
    sh[0] = reads[1];                  // stack_top
    sh[1] = ax + eqg(ax, 0.0f);        // ax_safe
  }
  __syncthreads();

  // ---- Phase 2: div enumeration (256 q across block), shifts (32 lanes) ----
  {
    const float a = sh[0], b = sh[1];
    const float q = (float)t;
    float t1 = a - q * b + 0.5f;
    float th1 = (silu_(SCL*(t1+0.5f)) - silu_(SCL*(t1-0.5f))) * (1.0f/SCL);
    float t2 = a - (q + 1.0f) * b + 0.5f;
    float th2 = (silu_(SCL*(t2+0.5f)) - silu_(SCL*(t2-0.5f))) * (1.0f/SCL);
    atomicAdd(&acc[0], (th1 - th2) * q);
    if (t < 32){
      float fi = (float)t;
      float p2 = exp2f(fi);
      float d  = ax - fi;
      float pulse = smul(sth(d + 0.5f), sth(-d + 0.5f));
      atomicAdd(&acc[1], smul(sh[0], p2) * pulse);
      atomicAdd(&acc[2], floorf(sh[0] / p2) * pulse);
    }
  }
  __syncthreads();

  // ---- Phase 3: scalar VM update -------------------------------------------
  if (t == 0){
    const float stack_top = sh[0];
    const float ax_safe   = sh[1];
    const float instruction   = reads[0];
    const float li_result     = reads[2];     // lc_result identical
    const float bp_from_stack = reads[3];
    const float pc_from_stack = reads[4];

    float opcode = fmodf(instruction, 256.0f);
    if (opcode < 0.0f) opcode += 256.0f;      // jnp.remainder semantics
    const float imm = floorf(instruction * (1.0f/256.0f));
    const float pc_next = pc + 8.0f;

    const float div_result = acc[0];
    const float shl_result = acc[1];
    const float shr_result = acc[2];

    const float add_result = stack_top + ax;
    const float sub_result = stack_top - ax;
    const float mul_result = smul(stack_top, ax);
    const float mod_result = stack_top - smul(div_result, ax_safe);
    const float or_result  = stack_top + ax -        mul_result;
    const float xor_result = stack_top + ax - 2.0f * mul_result;
    const float and_result = mul_result;
    const float eq_result  = eqg(stack_top, ax);
    const float ne_result  = 1.0f - eq_result;
    const float lt_result  = gtg(ax, stack_top);
    const float gt_result  = gtg(stack_top, ax);
    const float le_result  = geg(ax, stack_top);
    const float ge_result  = geg(stack_top, ax);
    const float lea_result = bp + imm;

    // opcode gates: LEA=0 IMM=1 JMP=2 JSR=3 BZ=4 BNZ=5 ENT=6 ADJ=7 LEV=8
    // LI=9 LC=10 SI=11 SC=12 PSH=13 OR=14 XOR=15 AND=16 EQ=17 NE=18 LT=19
    // GT=20 LE=21 GE=22 SHL=23 SHR=24 ADD=25 SUB=26 MUL=27 DIV=28 MOD=29
    float g[30];
    #pragma unroll
    for (int o = 0; o < 30; ++o) g[o] = eqg(opcode, (float)o);

    const float gate_sum = g[0]+g[1]+g[9]+g[10]+g[25]+g[26]+g[27]+g[28]+g[29]
                         + g[23]+g[24]+g[14]+g[15]+g[16]+g[17]+g[18]+g[19]
                         + g[20]+g[21]+g[22];
    const float new_ax = ax * (1.0f - gate_sum)
      + lea_result*g[0] + imm*g[1] + li_result*g[9] + li_result*g[10]
      + add_result*g[25] + sub_result*g[26] + mul_result*g[27]
      + div_result*g[28] + mod_result*g[29] + shl_result*g[23]
      + shr_result*g[24] + or_result*g[14] + xor_result*g[15]
      + and_result*g[16] + eq_result*g[17] + ne_result*g[18]
      + lt_result*g[19] + gt_result*g[20] + le_result*g[21] + ge_result*g[22];

    const float pops = g[25]+g[26]+g[27]+g[28]+g[29]+g[23]+g[24]+g[14]+g[15]
                     + g[16]+g[17]+g[18]+g[19]+g[20]+g[21]+g[22];
    const float g_psh = g[13], g_adj = g[7], g_ent = g[6], g_lev = g[8];
    const float new_sp = sp*(1.0f - g_psh - g_adj - g_ent - g_lev - pops)
      + (sp-8.0f)*g_psh + (sp+imm)*g_adj + (sp-imm)*g_ent
      + bp*g_lev + (sp+8.0f)*pops;

    const float new_bp = bp*(1.0f - g_ent - g_lev) + sp*g_ent
                       + bp_from_stack*g_lev;

    const float g_jmp = g[2], g_jsr = g[3], g_bz = g[4], g_bnz = g[5];
    const float eqax0 = eqg(ax, 0.0f);
    const float bz_take  = smul(g_bz, eqax0);
    const float bnz_take = smul(g_bnz, 1.0f - eqax0);
    const float new_pc = pc_next*(1.0f - g_jmp - g_jsr - bz_take - bnz_take - g_lev)
      + imm*g_jmp + imm*g_jsr + imm*bz_take + imm*bnz_take
      + pc_from_stack*g_lev;

    out[0] = rintf(new_pc);
    out[1] = rintf(new_sp);
    out[2] = rintf(new_bp);
    out[3] = rintf(new_ax);

    const float g_si = g[11], g_sc = g[12];
    ws[0] = (sp - 8.0f)*g_psh;  ws[5] = ax*g_psh;
    ws[1] = (sp - 8.0f)*g_jsr;  ws[6] = pc_next*g_jsr;
    ws[2] = (sp - 8.0f)*g_ent;  ws[7] = bp*g_ent;
    ws[3] = stack_top*g_si;     ws[8] = ax*g_si;
    ws[4] = stack_top*g_sc;     ws[9] = ax*g_sc;
  }
}

// ---------------------------------------------------------------------------
// Kernel B: bandwidth-bound streaming pass. out = mem with the 5 sequential
// soft writes applied; gates are exactly 0 in fp32 outside a ±9.5 window so
// the hot path is a pure float4 copy (128 MB of traffic).
// ---------------------------------------------------------------------------
__global__ __launch_bounds__(256)
void vm_mem_update(const float* __restrict__ mem, const float* __restrict__ ws,
                   float* __restrict__ outmem, int n)
{
  __shared__ float wa[5], wv[5];
  if (threadIdx.x < 5){
    wa[threadIdx.x] = ws[threadIdx.x];
    wv[threadIdx.x] = ws[5 + threadIdx.x];
  }
  __syncthreads();

  const long i0 = ((long)blockIdx.x * blockDim.x + threadIdx.x) * 4;
  if (i0 >= n) return;
  if (i0 + 32768 < n) __builtin_prefetch(mem + i0 + 32768, 0, 0);

  float4 v;
  if (i0 + 3 < n){
    v = *(const float4*)(mem + i0);
  } else {
    v.x = mem[i0]; v.y = 0.f; v.z = 0.f; v.w = 0.f;
    if (i0 + 1 < n) v.y = mem[i0 + 1];
    if (i0 + 2 < n) v.z = mem[i0 + 2];
  }

  const float base = (float)i0;        // < 2^24, exact in fp32
  bool touched = false;
  #pragma unroll
  for (int k = 0; k < 5; ++k){
    float a = wa[k];
    touched |= (a > base - 9.6f) && (a < base + 12.6f);
  }

  if (touched){
    float e[4] = {v.x, v.y, v.z, v.w};
    #pragma unroll
    for (int j = 0; j < 4; ++j){
      float pos = base + (float)j;
      float m = e[j];
      #pragma unroll
      for (int k = 0; k < 5; ++k){     // sequential composition, ref order
        float d = pos - wa[k];
        if (fabsf(d) < 9.6f){
          float gg = eqg(wa[k], pos);
          m = m * (1.0f - gg) + wv[k] * gg;
        }
      }
      e[j] = m;
    }
    v = make_float4(e[0], e[1], e[2], e[3]);
  }

  if (i0 + 3 < n){
    *(float4*)(outmem + i0) = v;
  } else {
    outmem[i0] = v.x;
    if (i0 + 1 < n) outmem[i0 + 1] = v.y;
    if (i0 + 2 < n) outmem[i0 + 2] = v.z;
  }
}

extern "C" void kernel_launch(void* const* d_in, const int* in_sizes, int n_in,
                              void* d_out, int out_size, void* d_ws, size_t ws_size,
                              hipStream_t stream)
{
  const float* pc  = (const float*)d_in[0];
  const float* sp  = (const float*)d_in[1];
  const float* bp  = (const float*)d_in[2];
  const float* ax  = (const float*)d_in[3];
  const float* mem = (const float*)d_in[4];
  const int n = in_sizes[4];

  float* out = (float*)d_out;
  float* ws  = (float*)d_ws;

  vm_step_scalar<<<1, 256, 0, stream>>>(pc, sp, bp, ax, mem, out, ws);

  const long elems4 = ((long)n + 3) / 4;
  const int blocks = (int)((elems4 + 255) / 256);
  vm_mem_update<<<blocks, 256, 0, stream>>>(mem, ws, out + 4, n);
}